// CDKVMN_627065225819
// MI455X (gfx1250) — compile-verified
//
#include <hip/hip_runtime.h>
#include <hip/hip_bf16.h>
#include <math.h>

// ---------------------------------------------------------------------------
// DKVMN forward, fused single kernel for MI455X (gfx1250, wave32).
//   grid = 64 (one batch element per workgroup), block = 128 (4 waves)
//   GEMMs: v_wmma_f32_16x16x4_f32 with B matrices pre-swizzled into LDS in
//          fragment order (branch-free ds_load_b64 per K-step), A fragments
//          hoisted into VGPRs and reused across N-tiles / output matrices.
//   scan:  per-lane recurrence, Mv[50] in VGPRs, w/e/a staged in LDS,
//          no barriers inside the 200-step loop.
// ---------------------------------------------------------------------------

typedef __attribute__((ext_vector_type(2))) float v2f;
typedef __attribute__((ext_vector_type(8))) float v8f;

#define B_   64
#define L1_  199
#define L_   200
#define NC_  1000
#define D_   64
#define M_   50

#define NT_M 13   // row tiles: 200 -> 13*16 = 208
#define NT_N 4    // col tiles: 64

// LDS layout (floats). e_s is reused as read_s during/after the scan.
// Fragment region holds MkT/We/Wa frags in phase 1, Wf frags in phase 3.
#define OFF_W     0
#define OFF_E     (L_ * 64)
#define OFF_A     (2 * L_ * 64)
#define OFF_FRAG  (3 * L_ * 64)            // 12288 floats (3 x 4096)
#define OFF_P     (OFF_FRAG + 12288)       // 256 floats
#define SMEM_FLOATS (OFF_P + 256)          // 50944 floats ~ 199 KB

#define FRAG_MK   0                        // 2048 v2f slots each
#define FRAG_WE   2048
#define FRAG_WA   4096
#define FRAG_WF   0                        // 4096 v2f slots (phase 3)

__device__ __forceinline__ float sigmoidf_(float x) {
  return 1.0f / (1.0f + __expf(-x));
}

__device__ __forceinline__ v8f wmma4(v2f a, v2f b, v8f c) {
  // (neg_a, A, neg_b, B, c_mod, C, reuse_a, reuse_b)
  return __builtin_amdgcn_wmma_f32_16x16x4_f32(false, a, false, b, (short)0, c,
                                               false, false);
}

__global__ void dkvmn_fused_kernel(
    const int* __restrict__ cseqs, const int* __restrict__ rseqs,
    const int* __restrict__ shft_cseqs, const int* __restrict__ shft_rseqs,
    const float* __restrict__ kemb, const float* __restrict__ vemb,
    const float* __restrict__ Mk, const float* __restrict__ Mv0,
    const float* __restrict__ Wf, const float* __restrict__ bf,
    const float* __restrict__ We, const float* __restrict__ be,
    const float* __restrict__ Wa, const float* __restrict__ ba,
    const float* __restrict__ Wp, const float* __restrict__ bp,
    float* __restrict__ out) {
  extern __shared__ float smem[];
  float* w_s  = smem + OFF_W;    // [200][64] logits -> softmax weights
  float* e_s  = smem + OFF_E;    // [200][64] erase gates, later: read values
  float* a_s  = smem + OFF_A;    // [200][64] add vectors
  v2f*   frag = (v2f*)(smem + OFF_FRAG);   // B fragments, WMMA order
  float* p_s  = smem + OFF_P;    // [256] per-row f.Wp accumulators

  const int b    = blockIdx.x;
  const int tid  = threadIdx.x;
  const int wave = tid >> 5;
  const int lane = tid & 31;
  const int half = lane >> 4;    // K-half of the 16x16x4 fragment layout
  const int l    = lane & 15;

  if (tid == 0) {
    __builtin_prefetch(Mk, 0, 3);
    __builtin_prefetch(We, 0, 3);
    __builtin_prefetch(Wa, 0, 3);
    __builtin_prefetch(Wf, 0, 3);
  }

  const int* cs  = cseqs + b * L1_;
  const int* rs  = rseqs + b * L1_;
  const int* scs = shft_cseqs + b * L1_;
  const int* srs = shft_rseqs + b * L1_;

  auto qval = [&](int t) -> int { return (t == 0) ? cs[0] : scs[t - 1]; };
  auto rval = [&](int t) -> int { return (t == 0) ? rs[0] : srs[t - 1]; };

  // ---- Phase 0: stage B matrices into LDS in WMMA-fragment order.
  // Slot s -> (nt = s>>9, kc = (s>>5)&15, lane_ = s&31); B[brow..brow+1][col]
  for (int s = tid; s < 2048; s += 128) {
    const int lane_ = s & 31, kc = (s >> 5) & 15, nt = s >> 9;
    const int brow = kc * 4 + 2 * (lane_ >> 4);
    const int col  = nt * 16 + (lane_ & 15);
    v2f mk, we, wa;
    if (col < M_) {                       // B = Mk^T, zero-pad cols 50..63
      mk = *(const v2f*)(Mk + col * D_ + brow);
    } else {
      mk[0] = 0.f; mk[1] = 0.f;
    }
    we[0] = We[brow * D_ + col];
    we[1] = We[(brow + 1) * D_ + col];
    wa[0] = Wa[brow * D_ + col];
    wa[1] = Wa[(brow + 1) * D_ + col];
    frag[FRAG_MK + s] = mk;
    frag[FRAG_WE + s] = we;
    frag[FRAG_WA + s] = wa;
  }
  __syncthreads();

  // ---- Phase 1: logits = k@Mk^T ; e = sigmoid(v@We+be) ; a = tanh(v@Wa+ba)
  for (int mt = wave; mt < NT_M; mt += 4) {
    const int m0   = mt * 16;
    const int row  = m0 + l;
    const int trow = (row < L_) ? row : (L_ - 1);  // clamp; stores skip >=200

    const int qi = qval(trow);
    const float* krow = kemb + qi * D_;
    const float* vrow = vemb + (qi + NC_ * rval(trow)) * D_;

    v2f Ak[16], Av[16];                  // hoisted A fragments (k and v rows)
#pragma unroll
    for (int kc = 0; kc < 16; ++kc) {
      Ak[kc] = *(const v2f*)(krow + kc * 4 + 2 * half);
      Av[kc] = *(const v2f*)(vrow + kc * 4 + 2 * half);
    }

    for (int nt = 0; nt < NT_N; ++nt) {
      const int col = nt * 16 + l;
      v8f accW = {0.f, 0.f, 0.f, 0.f, 0.f, 0.f, 0.f, 0.f};
      v8f accE = accW, accA = accW;
#pragma unroll
      for (int kc = 0; kc < 16; ++kc) {
        const int fi = (nt * 16 + kc) * 32 + lane;
        accW = wmma4(Ak[kc], frag[FRAG_MK + fi], accW);
        accE = wmma4(Av[kc], frag[FRAG_WE + fi], accE);
        accA = wmma4(Av[kc], frag[FRAG_WA + fi], accA);
      }
      const float beC = be[col];
      const float baC = ba[col];
#pragma unroll
      for (int i = 0; i < 8; ++i) {
        const int r = m0 + i + 8 * half;
        if (r < L_) {
          w_s[r * 64 + col] = accW[i];
          e_s[r * 64 + col] = sigmoidf_(accE[i] + beC);
          a_s[r * 64 + col] = tanhf(accA[i] + baC);
        }
      }
    }
  }
  __syncthreads();

  // ---- softmax over the 50 valid memory slots, per timestep row
  for (int t = tid; t < L_; t += 128) {
    float* row = w_s + t * 64;
    float mx = row[0];
    for (int m = 1; m < M_; ++m) mx = fmaxf(mx, row[m]);
    float s = 0.f;
    for (int m = 0; m < M_; ++m) {
      const float ex = __expf(row[m] - mx);
      row[m] = ex;
      s += ex;
    }
    const float inv = 1.0f / s;
    for (int m = 0; m < M_; ++m) row[m] *= inv;
  }
  __syncthreads();

  // ---- Phase 2: sequential scan. Lane d owns Mv[:, d] in 50 VGPRs.
  if (tid < D_) {
    const int d = tid;
    float Mv[M_];
#pragma unroll
    for (int m = 0; m < M_; ++m) Mv[m] = Mv0[m * D_ + d];

    for (int t = 0; t < L_; ++t) {
      const float et = e_s[t * 64 + d];
      const float at = a_s[t * 64 + d];
      float rd = 0.f;
#pragma unroll
      for (int mb = 0; mb < 12; ++mb) {
        const float4 w4 = *(const float4*)(w_s + t * 64 + 4 * mb);  // ds b128
        const float wv[4] = {w4.x, w4.y, w4.z, w4.w};
#pragma unroll
        for (int j = 0; j < 4; ++j) {
          const int m = 4 * mb + j;
          const float wm = wv[j];
          rd    = fmaf(wm, Mv[m], rd);                    // read (pre-update)
          Mv[m] = fmaf(wm, fmaf(-et, Mv[m], at), Mv[m]);  // erase + add
        }
      }
      {
        const float2 w2 = *(const float2*)(w_s + t * 64 + 48);
        const float wv[2] = {w2.x, w2.y};
#pragma unroll
        for (int j = 0; j < 2; ++j) {
          const int m = 48 + j;
          const float wm = wv[j];
          rd    = fmaf(wm, Mv[m], rd);
          Mv[m] = fmaf(wm, fmaf(-et, Mv[m], at), Mv[m]);
        }
      }
      e_s[t * 64 + d] = rd;   // e[t] is dead now; slot becomes read[t]
    }
  }
  __syncthreads();

  // ---- Stage Wf fragments (K=128 -> 32 k-chunks) into the frag region,
  //      and zero the p accumulators.
  for (int s = tid; s < 4096; s += 128) {
    const int lane_ = s & 31, kc = (s >> 5) & 31, nt = s >> 10;
    const int brow = kc * 4 + 2 * (lane_ >> 4);
    const int col  = nt * 16 + (lane_ & 15);
    v2f wf;
    wf[0] = Wf[brow * D_ + col];
    wf[1] = Wf[(brow + 1) * D_ + col];
    frag[FRAG_WF + s] = wf;
  }
  for (int i = tid; i < 256; i += 128) p_s[i] = 0.f;
  __syncthreads();

  // ---- Phase 3: f = tanh([read|k] @ Wf + bf) ; p = sigmoid(f.Wp + bp)
  const float* re_s = e_s;   // read values [200][64]
  for (int mt = wave; mt < NT_M; mt += 4) {
    const int m0   = mt * 16;
    const int row  = m0 + l;
    const int trow = (row < L_) ? row : (L_ - 1);

    const float* krow = kemb + qval(trow) * D_;

    v2f Ar[16], Ak2[16];
#pragma unroll
    for (int kc = 0; kc < 16; ++kc) {
      Ar[kc]  = *(const v2f*)(re_s + trow * 64 + kc * 4 + 2 * half);  // LDS
      Ak2[kc] = *(const v2f*)(krow + kc * 4 + 2 * half);              // global
    }

    for (int nt = 0; nt < NT_N; ++nt) {
      const int col = nt * 16 + l;
      v8f acc = {0.f, 0.f, 0.f, 0.f, 0.f, 0.f, 0.f, 0.f};
#pragma unroll
      for (int kc = 0; kc < 16; ++kc) {    // rows 0..63 of Wf (read part)
        acc = wmma4(Ar[kc], frag[FRAG_WF + (nt * 32 + kc) * 32 + lane], acc);
      }
#pragma unroll
      for (int kc = 0; kc < 16; ++kc) {    // rows 64..127 of Wf (k part)
        acc = wmma4(Ak2[kc],
                    frag[FRAG_WF + (nt * 32 + 16 + kc) * 32 + lane], acc);
      }
      const float bfc = bf[col];
      const float wpc = Wp[col];
#pragma unroll
      for (int i = 0; i < 8; ++i) {
        const int r = m0 + i + 8 * half;
        if (r < L_) {
          const float fv = tanhf(acc[i] + bfc);
          atomicAdd(&p_s[r], fv * wpc);    // ds_add_f32
        }
      }
    }
  }
  __syncthreads();

  const float bpv = bp[0];
  for (int t = tid; t < L_; t += 128) {
    out[b * L_ + t] = sigmoidf_(p_s[t] + bpv);
  }
}

extern "C" void kernel_launch(void* const* d_in, const int* in_sizes, int n_in,
                              void* d_out, int out_size, void* d_ws,
                              size_t ws_size, hipStream_t stream) {
  (void)in_sizes; (void)n_in; (void)out_size; (void)d_ws; (void)ws_size;

  const int*   cseqs      = (const int*)d_in[0];
  const int*   rseqs      = (const int*)d_in[1];
  const int*   shft_cseqs = (const int*)d_in[2];
  const int*   shft_rseqs = (const int*)d_in[3];
  const float* kemb       = (const float*)d_in[4];
  const float* vemb       = (const float*)d_in[5];
  const float* Mk         = (const float*)d_in[6];
  const float* Mv0        = (const float*)d_in[7];
  const float* Wf         = (const float*)d_in[8];
  const float* bf         = (const float*)d_in[9];
  const float* We         = (const float*)d_in[10];
  const float* be         = (const float*)d_in[11];
  const float* Wa         = (const float*)d_in[12];
  const float* ba         = (const float*)d_in[13];
  const float* Wp         = (const float*)d_in[14];
  const float* bp         = (const float*)d_in[15];
  float* out = (float*)d_out;

  const size_t smem_bytes = SMEM_FLOATS * sizeof(float);   // ~199 KB
  hipFuncSetAttribute(reinterpret_cast<const void*>(dkvmn_fused_kernel),
                      hipFuncAttributeMaxDynamicSharedMemorySize,
                      (int)smem_bytes);

  dkvmn_fused_kernel<<<B_, 128, smem_bytes, stream>>>(
      cseqs, rseqs, shft_cseqs, shft_rseqs, kemb, vemb, Mk, Mv0, Wf, bf, We,
      be, Wa, ba, Wp, bp, out);
}